// MsimilarityHyperrelationNetwork_57552561766620
// MI455X (gfx1250) — compile-verified
//
#include <hip/hip_runtime.h>
#include <hip/hip_bf16.h>

typedef __attribute__((ext_vector_type(2))) float v2f;
typedef __attribute__((ext_vector_type(8))) float v8f;

#define NB 8
#define NC 1024
#define NHW 4096          // 64*64
#define SM_W 512

__device__ __forceinline__ float readlane_f(float v, int lane) {
    int i = __builtin_bit_cast(int, v);
    int r = __builtin_amdgcn_readlane(i, lane);
    return __builtin_bit_cast(float, r);
}

// Exact fp32 cross-lane (wave32) sum via V_WMMA_F32_16X16X4_F32 with a
// ones B-matrix. A (16x4 f32): lanes 0-15 hold K=0,1; lanes 16-31 hold K=2,3.
// With a = {v, 0}: D[m,n] = v_m + v_{m+16}. Lane L<16 holds D[0..7][L] in the
// 8 accumulator VGPRs; lane L>=16 holds D[8..15][L-16]. Summing a lane's 8
// regs then readlane(0)+readlane(16) gives the 32-lane total, all in fp32.
__device__ __forceinline__ float wave_reduce_wmma(float v) {
    v2f a; a.x = v;    a.y = 0.0f;
    v2f b; b.x = 1.0f; b.y = 1.0f;
    v8f c = {};
    c = __builtin_amdgcn_wmma_f32_16x16x4_f32(
        /*neg_a=*/false, a, /*neg_b=*/false, b,
        /*c_mod=*/(short)0, c, /*reuse_a=*/false, /*reuse_b=*/false);
    float p = c[0] + c[1] + c[2] + c[3] + c[4] + c[5] + c[6] + c[7];
    return readlane_f(p, 0) + readlane_f(p, 16);
}

// Stage A: bilinear (align_corners) sample of the 512x512 mask at the
// position of row/col (yy,xx) of the virtual 128x128 intermediate grid.
__device__ __forceinline__ float sampleA(const float* __restrict__ sm, float yy, float xx) {
    const float stepA = 511.0f / 127.0f;
    float ys = yy * stepA;
    float xs = xx * stepA;
    int y0 = min((int)floorf(ys), 511);
    int x0 = min((int)floorf(xs), 511);
    int y1 = min(y0 + 1, 511);
    int x1 = min(x0 + 1, 511);
    float wy = ys - (float)y0;
    float wx = xs - (float)x0;
    float tl = sm[y0 * SM_W + x0];
    float tr = sm[y0 * SM_W + x1];
    float bl = sm[y1 * SM_W + x0];
    float br = sm[y1 * SM_W + x1];
    float top = tl * (1.0f - wx) + tr * wx;
    float bot = bl * (1.0f - wx) + br * wx;
    return top * (1.0f - wy) + bot * wy;
}

// Stage B on top of stage A: value of the final 64x64 mask at (y,x).
__device__ __forceinline__ float maskVal(const float* __restrict__ sm, int y, int x) {
    const float stepB = 127.0f / 63.0f;
    float ys = (float)y * stepB;
    float xs = (float)x * stepB;
    int y0 = min((int)floorf(ys), 127);
    int x0 = min((int)floorf(xs), 127);
    int y1 = min(y0 + 1, 127);
    int x1 = min(x0 + 1, 127);
    float wy = ys - (float)y0;
    float wx = xs - (float)x0;
    float tl = sampleA(sm, (float)y0, (float)x0);
    float tr = sampleA(sm, (float)y0, (float)x1);
    float bl = sampleA(sm, (float)y1, (float)x0);
    float br = sampleA(sm, (float)y1, (float)x1);
    float top = tl * (1.0f - wx) + tr * wx;
    float bot = bl * (1.0f - wx) + br * wx;
    return top * (1.0f - wy) + bot * wy;
}

// One block per batch. Computes the 64x64 mask, the positive-count and
// positive-sum (WMMA wave reductions + tiny LDS combine), the mean threshold,
// then writes sel (0/1 floats) and the fused multiplier mask*sel.
__global__ __launch_bounds__(256) void mask_mean_sel_kernel(
    const float* __restrict__ support_mask,
    float* __restrict__ mult,      // [NB, NHW] workspace
    float* __restrict__ sel_out) { // [NB, NHW] second tuple output
    __shared__ float sMask[NHW];
    __shared__ float sCnt[8];
    __shared__ float sSum[8];
    __shared__ float sMean;

    const int b = blockIdx.x;
    const int tid = threadIdx.x;
    const float* smb = support_mask + (size_t)b * SM_W * SM_W;

    float cnt = 0.0f, sum = 0.0f;
#pragma unroll
    for (int k = 0; k < 16; ++k) {
        int j = tid + 256 * k;
        int y = j >> 6;
        int x = j & 63;
        float m = maskVal(smb, y, x);
        sMask[j] = m;
        if (m > 0.0f) { cnt += 1.0f; sum += m; }
    }

    // Uniform control flow here: EXEC is all-1s for the WMMA reductions.
    float wc = wave_reduce_wmma(cnt);
    float ws = wave_reduce_wmma(sum);
    int wave = tid >> 5;
    if ((tid & 31) == 0) { sCnt[wave] = wc; sSum[wave] = ws; }
    __syncthreads();
    if (tid == 0) {
        float tc = 0.0f, tsum = 0.0f;
#pragma unroll
        for (int i = 0; i < 8; ++i) { tc += sCnt[i]; tsum += sSum[i]; }
        sMean = tsum / fmaxf(tc, 1.0f);
    }
    __syncthreads();
    const float mean = sMean;

#pragma unroll
    for (int k = 0; k < 16; ++k) {
        int j = tid + 256 * k;
        float m = sMask[j];
        float s = (m >= mean) ? 1.0f : 0.0f;
        size_t o = (size_t)b * NHW + j;
        sel_out[o] = s;
        mult[o] = m * s;
    }
}

// Streaming HBM-bound kernel: out = feature * (mask*sel) broadcast over C.
// 8*1024*4096 fp32 = 8,388,608 float4s; each thread handles 4 coalesced float4s.
__global__ __launch_bounds__(256) void apply_kernel(
    const float4* __restrict__ feat4,
    const float4* __restrict__ mult4, // [NB, NHW/4]
    float4* __restrict__ out4) {
#pragma unroll
    for (int i = 0; i < 4; ++i) {
        size_t idx = (size_t)blockIdx.x * 1024 + (size_t)threadIdx.x + (size_t)i * 256;
        size_t b = idx >> 20;          // 1024*4096/4 = 2^20 float4s per batch
        size_t j4 = idx & 1023;        // float4 index within a 4096-elem row
        float4 fv = feat4[idx];
        float4 mv = mult4[b * (NHW / 4) + j4];
        float4 o;
        o.x = fv.x * mv.x;
        o.y = fv.y * mv.y;
        o.z = fv.z * mv.z;
        o.w = fv.w * mv.w;
        out4[idx] = o;
    }
}

extern "C" void kernel_launch(void* const* d_in, const int* in_sizes, int n_in,
                              void* d_out, int out_size, void* d_ws, size_t ws_size,
                              hipStream_t stream) {
    const float* feature      = (const float*)d_in[0]; // [8,1024,64,64]
    const float* support_mask = (const float*)d_in[1]; // [8,512,512]

    float* out = (float*)d_out;                               // [8,1024,4096]
    float* sel_out = out + (size_t)NB * NC * NHW;             // [8,4096] as 0/1 floats
    float* mult = (float*)d_ws;                               // [8,4096] scratch

    mask_mean_sel_kernel<<<NB, 256, 0, stream>>>(support_mask, mult, sel_out);

    const size_t total4 = (size_t)NB * NC * NHW / 4;          // 8,388,608
    const int blocks = (int)(total4 / 1024);                  // 8192
    apply_kernel<<<blocks, 256, 0, stream>>>(
        (const float4*)feature, (const float4*)mult, (float4*)out);
}